// GenerativeInfoNCELoss_26508538151633
// MI455X (gfx1250) — compile-verified
//
#include <hip/hip_runtime.h>
#include <hip/hip_bf16.h>

typedef __attribute__((ext_vector_type(2))) float v2f;
typedef __attribute__((ext_vector_type(8))) float v8f;

namespace {
constexpr int Bc   = 32;
constexpr int Sc   = 512;
constexpr int Hc   = 1024;
constexpr int NEGc = 10;
constexpr int SM1  = Sc - 1;               // 511 prediction rows per batch
constexpr int TILES = (SM1 + 15) / 16;     // 32 tiles of 16 rows per batch
constexpr int WAVES_PER_BLOCK = 4;
}

// dynamic extract from an 8-wide f32 accumulator (lowers to v_cndmask chain)
__device__ __forceinline__ float pick8(v8f v, int r) {
  float x = v[0];
  if (r == 1) x = v[1];
  if (r == 2) x = v[2];
  if (r == 3) x = v[3];
  if (r == 4) x = v[4];
  if (r == 5) x = v[5];
  if (r == 6) x = v[6];
  if (r == 7) x = v[7];
  return x;
}

// Load a 2-float chunk at (uniform base) + (per-lane 32-bit byte offset) + k
// byte offset. 32-bit voffset + scalar base encourages the GLOBAL SADDR form.
__device__ __forceinline__ v2f ld2(const float* __restrict__ base,
                                   unsigned off, int kbytes) {
  return *(const v2f*)((const char*)base + (size_t)off + (size_t)kbytes);
}

__global__ __launch_bounds__(WAVES_PER_BLOCK * 32)
void infonce_wmma_kernel(const float* __restrict__ enc,
                         const float* __restrict__ evt,
                         const int*   __restrict__ negidx,
                         const float* __restrict__ temp,
                         float*       __restrict__ out) {
  const int wave   = threadIdx.x >> 5;
  const int lane   = threadIdx.x & 31;
  const int tileId = blockIdx.x * WAVES_PER_BLOCK + wave;   // 0 .. 1023
  const int b      = tileId / TILES;
  const int s0     = (tileId % TILES) * 16;

  // 32-bit f32 WMMA operand layout (A 16x4 and B(=X^T) 4x16 identical in regs):
  // lane L<16 -> row L, elems [k, k+1]; lane L>=16 -> row L-16, elems [k+2, k+3]
  const int m     = lane & 15;
  const unsigned koffB = (unsigned)((lane >> 4) << 3);   // 0 or 8 bytes

  const int s  = s0 + m;
  const int sc = (s < SM1) ? s : (SM1 - 1);              // clamp pad rows (discarded)

  // Per-lane 32-bit byte offsets (arrays are 64MB -> fits u32), bases uniform.
  const unsigned offP = (unsigned)(b * Sc + sc) * (Hc * 4u) + koffB;      // in enc
  const unsigned offQ = (unsigned)(b * Sc + sc + 1) * (Hc * 4u) + koffB;  // in evt
  unsigned offN[NEGc];
#pragma unroll
  for (int j = 0; j < NEGc; ++j) {
    const int idx = negidx[(size_t)(b * SM1 + sc) * NEGc + j];
    const int g   = idx + ((idx >= b * Sc) ? Sc : 0);    // skip own batch block
    offN[j] = (unsigned)g * (Hc * 4u) + koffB;
  }

  v8f acc_pq = {}, acc_pp = {}, acc_qq = {};
  v8f acc_n[NEGc];
#pragma unroll
  for (int j = 0; j < NEGc; ++j) acc_n[j] = acc_pq;

  // ---- software pipeline: prologue loads for chunk k=0 ----
  v2f cA = ld2(enc, offP, 0);
  v2f cQ = ld2(evt, offQ, 0);
  v2f cN[NEGc];
#pragma unroll
  for (int j = 0; j < NEGc; ++j) cN[j] = ld2(evt, offN[j], 0);

#pragma unroll 2
  for (int k = 16; k <= (Hc - 4) * 4; k += 16) {   // k in bytes, chunk stride 16B
    // issue next chunk's 12 loads (independent of current WMMAs)
    v2f nA = ld2(enc, offP, k);
    v2f nQ = ld2(evt, offQ, k);
    v2f nN[NEGc];
#pragma unroll
    for (int j = 0; j < NEGc; ++j) nN[j] = ld2(evt, offN[j], k);

    // consume current chunk
    acc_pq = __builtin_amdgcn_wmma_f32_16x16x4_f32(false, cA, false, cQ, (short)0, acc_pq, false, false);
    acc_pp = __builtin_amdgcn_wmma_f32_16x16x4_f32(false, cA, false, cA, (short)0, acc_pp, false, false);
    acc_qq = __builtin_amdgcn_wmma_f32_16x16x4_f32(false, cQ, false, cQ, (short)0, acc_qq, false, false);
#pragma unroll
    for (int j = 0; j < NEGc; ++j)
      acc_n[j] = __builtin_amdgcn_wmma_f32_16x16x4_f32(false, cA, false, cN[j], (short)0, acc_n[j], false, false);

    cA = nA; cQ = nQ;
#pragma unroll
    for (int j = 0; j < NEGc; ++j) cN[j] = nN[j];
  }

  // ---- epilogue: last chunk ----
  acc_pq = __builtin_amdgcn_wmma_f32_16x16x4_f32(false, cA, false, cQ, (short)0, acc_pq, false, false);
  acc_pp = __builtin_amdgcn_wmma_f32_16x16x4_f32(false, cA, false, cA, (short)0, acc_pp, false, false);
  acc_qq = __builtin_amdgcn_wmma_f32_16x16x4_f32(false, cQ, false, cQ, (short)0, acc_qq, false, false);
#pragma unroll
  for (int j = 0; j < NEGc; ++j)
    acc_n[j] = __builtin_amdgcn_wmma_f32_16x16x4_f32(false, cA, false, cN[j], (short)0, acc_n[j], false, false);

  // Diagonal D[m][m] ownership: m<8 -> lane m, vgpr m; m>=8 -> lane m+16, vgpr m-8.
  float contrib = 0.0f;
  int r = -1, mr = 0;
  if (lane < 8)        { r = lane;      mr = lane;      }
  else if (lane >= 24) { r = lane - 24; mr = lane - 16; }

  if (r >= 0 && (s0 + mr) < SM1) {
    const float invT = 1.0f / temp[0];
    const float pq = pick8(acc_pq, r);
    const float pp = pick8(acc_pp, r);
    const float qq = pick8(acc_qq, r);
    const float pn = fmaxf(sqrtf(pp), 1e-8f);
    const float qn = fmaxf(sqrtf(qq), 1e-8f);
    const float l0 = pq / (pn * qn) * invT;
    float ln[NEGc];
    float mx = l0;
#pragma unroll
    for (int j = 0; j < NEGc; ++j) {
      ln[j] = pick8(acc_n[j], r) * invT;
      mx = fmaxf(mx, ln[j]);
    }
    float se = __expf(l0 - mx);
#pragma unroll
    for (int j = 0; j < NEGc; ++j) se += __expf(ln[j] - mx);
    contrib = (mx + __logf(se)) - l0;          // logsumexp - logits[...,0]
  }

  // wave32 tree reduction, then one atomic per wave (pre-scaled mean)
#pragma unroll
  for (int off = 16; off > 0; off >>= 1)
    contrib += __shfl_down(contrib, off, 32);
  if (lane == 0)
    atomicAdd(out, contrib * (1.0f / (float)(Bc * SM1)));
}

extern "C" void kernel_launch(void* const* d_in, const int* in_sizes, int n_in,
                              void* d_out, int out_size, void* d_ws, size_t ws_size,
                              hipStream_t stream) {
  const float* enc    = (const float*)d_in[0];
  const float* evt    = (const float*)d_in[1];
  const int*   negidx = (const int*)d_in[2];
  const float* temp   = (const float*)d_in[3];
  float* out = (float*)d_out;

  hipMemsetAsync(out, 0, sizeof(float), stream);   // capture-safe

  const int total_tiles = Bc * TILES;              // 1024 waves
  dim3 grid(total_tiles / WAVES_PER_BLOCK);        // 256 blocks
  dim3 block(WAVES_PER_BLOCK * 32);                // 4 waves / block
  hipLaunchKernelGGL(infonce_wmma_kernel, grid, block, 0, stream,
                     enc, evt, negidx, temp, out);
}